// dressed_quantum_circuit_75376676045298
// MI455X (gfx1250) — compile-verified
//
#include <hip/hip_runtime.h>

#define N_QUBITS 4
#define Q_DEPTH  6
#define IN_DIM   768
#define OUT_DIM  32
#define ROWS_PER_BLOCK 256

typedef __attribute__((ext_vector_type(2))) float v2f;
typedef __attribute__((ext_vector_type(8))) float v8f;

// RY(theta) on qubit w (bit weight 8>>w in C-order flatten of [2,2,2,2]).
__device__ __forceinline__ void ry_gate(float st[16], int w, float c, float s) {
    const int tb = 8 >> w;
    #pragma unroll
    for (int i = 0; i < 16; ++i) {
        if ((i & tb) == 0) {
            float a0 = st[i], a1 = st[i | tb];
            st[i]      = c * a0 - s * a1;
            st[i | tb] = s * a0 + c * a1;
        }
    }
}

// CNOT(control cwq, target twq): swap target bit within control=1 subspace.
__device__ __forceinline__ void cnot_gate(float st[16], int cwq, int twq) {
    const int cb = 8 >> cwq, tb = 8 >> twq;
    #pragma unroll
    for (int i = 0; i < 16; ++i) {
        if ((i & cb) != 0 && (i & tb) == 0) {
            float t = st[i]; st[i] = st[i | tb]; st[i | tb] = t;
        }
    }
}

__global__ __launch_bounds__(256) void dressed_qc_fused(
    const float* __restrict__ inputs,   // [B, 768]
    const float* __restrict__ W1,       // [768, 4]
    const float* __restrict__ b1,       // [4]
    const float* __restrict__ qw,       // [6, 4]
    const float* __restrict__ W2,       // [4, 32]
    const float* __restrict__ b2,       // [32]
    float* __restrict__ out)            // [B, 32]
{
    // Zero-padded, K-pair-interleaved W1 for direct WMMA B-fragment loads:
    //   w1kp[kp][n][j] = (n<4) ? W1[(2*kp+j)*4 + n] : 0   (kp = k/2, j = 0..1)
    // Lane (h, n) reads its B fragment {K=k0+2h, K=k0+2h+1} as ONE ds_load_b64.
    __shared__ float w1kp[(IN_DIM / 2) * 32];       // 384*32 floats = 48KB
    __shared__ float cwt[Q_DEPTH * N_QUBITS];       // cos(qw/2)
    __shared__ float swt[Q_DEPTH * N_QUBITS];       // sin(qw/2)
    __shared__ float xang[ROWS_PER_BLOCK * 4];      // pre-activation dot products

    const int tid  = threadIdx.x;
    const int lane = tid & 31;
    const int wv   = tid >> 5;                      // wave id 0..7
    const int blockRow = blockIdx.x * ROWS_PER_BLOCK;

    // ---- build padded B table + weight trig tables (one-time per block) ----
    #pragma unroll
    for (int t = 0; t < (IN_DIM / 2) * 32 / ROWS_PER_BLOCK; ++t) {
        const int idx = tid + t * ROWS_PER_BLOCK;
        const int kp  = idx >> 5;
        const int r   = idx & 31;
        const int nn  = r >> 1;
        const int j   = r & 1;
        w1kp[idx] = (nn < N_QUBITS) ? W1[(2 * kp + j) * N_QUBITS + nn] : 0.0f;
    }
    if (tid < Q_DEPTH * N_QUBITS) {
        float h = 0.5f * qw[tid];
        cwt[tid] = cosf(h);
        swt[tid] = sinf(h);
    }
    __syncthreads();

    // ---- GEMM1 via fp32 WMMA: two 16-row tiles per wave, shared B fragment ----
    const int n = lane & 15;             // A: row M = n ; B/D: col N = n
    const int h = lane >> 4;             // half-wave selector: A K-offset = 2h

    const size_t rowA0 = (size_t)(blockRow + wv * 32 + n);
    const float* pA0 = inputs + rowA0 * IN_DIM + 2 * h;
    const float* pA1 = pA0 + (size_t)16 * IN_DIM;
    const float* pB  = w1kp + h * 32 + 2 * n;       // + (k0/2)*32 per step

    v8f acc0 = {};
    v8f acc1 = {};
    #pragma unroll 4
    for (int k0 = 0; k0 < IN_DIM; k0 += 4) {
        v2f a0 = *(const v2f*)(pA0 + k0);
        v2f a1 = *(const v2f*)(pA1 + k0);
        v2f b  = *(const v2f*)(pB + (k0 >> 1) * 32);
        acc0 = __builtin_amdgcn_wmma_f32_16x16x4_f32(false, a0, false, b,
                                                     (short)0, acc0, false, false);
        acc1 = __builtin_amdgcn_wmma_f32_16x16x4_f32(false, a1, false, b,
                                                     (short)0, acc1, false, false);
    }

    // ---- scatter D columns 0..3 to LDS (C/D layout: VGPR j <-> M = j + 8h) ----
    if (n < N_QUBITS) {
        const int rb = wv * 32 + 8 * h;
        #pragma unroll
        for (int j = 0; j < 8; ++j) {
            xang[(rb + j) * 4 + n]      = acc0[j];
            xang[(rb + 16 + j) * 4 + n] = acc1[j];
        }
    }
    __syncthreads();

    // ---- per-thread: one row's quantum circuit, fully in registers ----
    float cc[4], ss[4];
    #pragma unroll
    for (int q = 0; q < 4; ++q) {
        float t  = tanhf(xang[tid * 4 + q] + b1[q]);
        float hh = t * 0.78539816339744830962f;      // (t * pi/2) * 0.5
        cc[q] = cosf(hh);
        ss[q] = sinf(hh);
    }

    float st[16];
    #pragma unroll
    for (int i = 0; i < 16; ++i) st[i] = 0.25f;      // H layer on |0000>

    #pragma unroll
    for (int w = 0; w < 4; ++w) ry_gate(st, w, cc[w], ss[w]);

    #pragma unroll
    for (int k = 0; k < Q_DEPTH; ++k) {
        cnot_gate(st, 0, 1);
        cnot_gate(st, 2, 3);
        cnot_gate(st, 1, 2);
        #pragma unroll
        for (int w = 0; w < 4; ++w) ry_gate(st, w, cwt[k * 4 + w], swt[k * 4 + w]);
    }

    float p[16];
    #pragma unroll
    for (int i = 0; i < 16; ++i) p[i] = st[i] * st[i];
    float z[4];
    #pragma unroll
    for (int w = 0; w < 4; ++w) {
        const int tb = 8 >> w;
        float a = 0.0f;
        #pragma unroll
        for (int i = 0; i < 16; ++i) a += (i & tb) ? -p[i] : p[i];
        z[w] = a;
    }

    // ---- q @ W2 + b2, 128B float4 stores ----
    float* orow = out + (size_t)(blockRow + tid) * OUT_DIM;
    #pragma unroll
    for (int o4 = 0; o4 < OUT_DIM / 4; ++o4) {
        float4 v;
        float* vp = (float*)&v;
        #pragma unroll
        for (int j = 0; j < 4; ++j) {
            const int o = o4 * 4 + j;
            float a = b2[o];
            #pragma unroll
            for (int q = 0; q < 4; ++q) a += z[q] * W2[q * OUT_DIM + o];
            vp[j] = a;
        }
        *(float4*)(orow + o4 * 4) = v;
    }
}

extern "C" void kernel_launch(void* const* d_in, const int* in_sizes, int n_in,
                              void* d_out, int out_size, void* d_ws, size_t ws_size,
                              hipStream_t stream) {
    const float* inputs = (const float*)d_in[0];
    const float* W1     = (const float*)d_in[1];
    const float* b1     = (const float*)d_in[2];
    const float* qw     = (const float*)d_in[3];
    const float* W2     = (const float*)d_in[4];
    const float* b2     = (const float*)d_in[5];
    float* out = (float*)d_out;

    const int B = in_sizes[0] / IN_DIM;          // 262144
    const int blocks = B / ROWS_PER_BLOCK;       // 1024 (B divisible by 256)
    dressed_qc_fused<<<blocks, 256, 0, stream>>>(inputs, W1, b1, qw, W2, b2, out);
}